// FarthestSubsample_2869038153767
// MI455X (gfx1250) — compile-verified
//
#include <hip/hip_runtime.h>

#define B      16
#define N      8192
#define CC     3
#define CV     64
#define NP     4096          // round(N * 0.5)
#define BLOCK  1024
#define PPT    (N / BLOCK)   // 8 points per thread
#define NWAVES (BLOCK / 32)  // 32 waves (wave32)

#if __has_builtin(__builtin_amdgcn_global_load_async_to_lds_b128)
#define HAVE_ASYNC_LDS 1
#else
#define HAVE_ASYNC_LDS 0
#endif

typedef int v4i __attribute__((ext_vector_type(4)));
typedef __attribute__((address_space(1))) v4i Gv4i;   // global-memory v4i
typedef __attribute__((address_space(3))) v4i Lv4i;   // LDS v4i

// ---------------------------------------------------------------------------
// FPS kernel: one workgroup per batch. Coords staged to LDS via the CDNA5
// async data-mover path, per-point state held in VGPRs, per-step argmax via
// wave32 shuffles + one cross-wave LDS reduction. 2 barriers / step.
// ---------------------------------------------------------------------------
__global__ void __launch_bounds__(BLOCK)
fps_kernel(const float* __restrict__ coords, int* __restrict__ idx_out) {
    extern __shared__ float smem[];          // 3*N floats: x | y | z
    __shared__ float rdist[NWAVES];
    __shared__ int   ridx[NWAVES];
    __shared__ int   bidx;

    const int b = blockIdx.x;
    const int t = threadIdx.x;
    const float* cb = coords + (size_t)b * 3 * N;   // [3][N] channel-major

    // ---- Stage coords (96KB) global -> LDS with async b128 copies ----
    const int total4 = (3 * N) / 4;                 // 6144 float4 transfers
    for (int i = t; i < total4; i += BLOCK) {
#if HAVE_ASYNC_LDS
        __builtin_amdgcn_global_load_async_to_lds_b128(
            (Gv4i*)(cb + 4 * i),
            (Lv4i*)(smem + 4 * i),
            /*offset=*/0, /*cpol=*/0);
#else
        float4 v = ((const float4*)cb)[i];
        ((float4*)smem)[i] = v;
#endif
    }
#if HAVE_ASYNC_LDS
#if __has_builtin(__builtin_amdgcn_s_wait_asynccnt)
    __builtin_amdgcn_s_wait_asynccnt(0);
#else
    asm volatile("s_wait_asynccnt 0" ::: "memory");
#endif
#endif
    if (t == 0) { bidx = 0; idx_out[(size_t)b * NP] = 0; }
    __syncthreads();

    const float* sx = smem;
    const float* sy = smem + N;
    const float* sz = smem + 2 * N;

    // ---- Own points -> registers (strided ownership: n = t + j*BLOCK) ----
    float px[PPT], py[PPT], pz[PPT], dmin[PPT];
#pragma unroll
    for (int j = 0; j < PPT; ++j) {
        const int n = t + j * BLOCK;
        px[j] = sx[n]; py[j] = sy[n]; pz[j] = sz[n];
        dmin[j] = 1e10f;
    }

    // ---- 4095 sequential FPS rounds ----
    for (int s = 1; s < NP; ++s) {
        const int   ci = bidx;                 // LDS broadcast read
        const float cx = sx[ci], cy = sy[ci], cz = sz[ci];

        float bestd = -1.0f; int bestn = 0;
#pragma unroll
        for (int j = 0; j < PPT; ++j) {
            const float dx = px[j] - cx;
            const float dy = py[j] - cy;
            const float dz = pz[j] - cz;
            const float d  = dx * dx + dy * dy + dz * dz;
            const float m  = fminf(dmin[j], d);
            dmin[j] = m;
            if (m > bestd) { bestd = m; bestn = t + j * BLOCK; }
        }

        // wave32 argmax (prefer lower index on ties, like jnp.argmax)
#pragma unroll
        for (int off = 16; off >= 1; off >>= 1) {
            const float od = __shfl_xor(bestd, off, 32);
            const int   on = __shfl_xor(bestn, off, 32);
            if (od > bestd || (od == bestd && on < bestn)) { bestd = od; bestn = on; }
        }
        const int wave = t >> 5;
        if ((t & 31) == 0) { rdist[wave] = bestd; ridx[wave] = bestn; }
        __syncthreads();

        if (wave == 0) {                        // cross-wave argmax in wave 0
            float wd = rdist[t]; int wn = ridx[t];
#pragma unroll
            for (int off = 16; off >= 1; off >>= 1) {
                const float od = __shfl_xor(wd, off, 32);
                const int   on = __shfl_xor(wn, off, 32);
                if (od > wd || (od == wd && on < wn)) { wd = od; wn = on; }
            }
            if (t == 0) { bidx = wn; idx_out[(size_t)b * NP + s] = wn; }
        }
        __syncthreads();
    }
}

// ---------------------------------------------------------------------------
// Gather: out = [B,CC,NP] coords then [B,CV,NP] values, both gathered at idx.
// Pure memory pass (~50MB @ 23.3 TB/s). One thread per output element.
// ---------------------------------------------------------------------------
__global__ void gather_kernel(const float* __restrict__ coords,
                              const float* __restrict__ values,
                              const int*   __restrict__ idx,
                              float* __restrict__ out) {
    const size_t total = (size_t)B * (CC + CV) * NP;
    const size_t gid = (size_t)blockIdx.x * blockDim.x + threadIdx.x;
    if (gid >= total) return;

    const int k = (int)(gid % NP);
    const int c = (int)((gid / NP) % (CC + CV));
    const int b = (int)(gid / ((size_t)NP * (CC + CV)));
    const int n = idx[(size_t)b * NP + k];

    if (c < CC) {
        const float* src = coords + ((size_t)b * CC + c) * N;
        __builtin_prefetch(src + n, 0, 0);              // global_prefetch_b8
        out[((size_t)b * CC + c) * NP + k] = src[n];
    } else {
        const int cv = c - CC;
        const float* src = values + ((size_t)b * CV + cv) * N;
        __builtin_prefetch(src + n, 0, 0);
        out[(size_t)B * CC * NP + ((size_t)b * CV + cv) * NP + k] = src[n];
    }
}

// ---------------------------------------------------------------------------
extern "C" void kernel_launch(void* const* d_in, const int* in_sizes, int n_in,
                              void* d_out, int out_size, void* d_ws, size_t ws_size,
                              hipStream_t stream) {
    const float* coords = (const float*)d_in[0];   // [16, 3, 8192]
    const float* values = (const float*)d_in[1];   // [16, 64, 8192]
    int* idx = (int*)d_ws;                          // B*NP ints = 256KB scratch

    const size_t dynLds = (size_t)3 * N * sizeof(float);   // 96KB (<320KB/WGP)
    (void)hipFuncSetAttribute((const void*)fps_kernel,
                              hipFuncAttributeMaxDynamicSharedMemorySize,
                              (int)dynLds);

    fps_kernel<<<B, BLOCK, dynLds, stream>>>(coords, idx);

    const size_t total = (size_t)B * (CC + CV) * NP;
    const int threads = 256;
    const int blocks = (int)((total + threads - 1) / threads);
    gather_kernel<<<blocks, threads, 0, stream>>>(coords, values, idx, (float*)d_out);
}